// TempDisentangler_60181081751582
// MI455X (gfx1250) — compile-verified
//
#include <hip/hip_runtime.h>
#include <hip/hip_bf16.h>
#include <math.h>

// ---------------------------------------------------------------------------
// MI455X (gfx1250) implementation, fp32 throughout via V_WMMA_F32_16X16X4_F32.
//  - conv branch: 4 causal convs folded into one k=8 conv, 32x32-blocked GEMM
//  - attention:   flash attention with S^T/O^T register trick (wave32 layout)
//  - freq branch: rfft/irfft expressed as DFT GEMMs (freq axis padded to 544)
// Register blocking gives ~1 load per WMMA; x is pre-transposed through LDS
// so A-fragments load as global_load_b64.
// ---------------------------------------------------------------------------

typedef __attribute__((ext_vector_type(2))) float v2f;
typedef __attribute__((ext_vector_type(8))) float v8f;

#define B_   16
#define C_   256
#define T_   1024
#define H_   256
#define NF   513
#define NFP  544   // 513 padded to multiple of 32 (zero rows/cols)
#define DH   64
#define NHEADS 4

__device__ __forceinline__ v8f wmma4(v2f a, v2f b, v8f c) {
  // D = A(16x4 f32) * B(4x16 f32) + C(16x16 f32)
  return __builtin_amdgcn_wmma_f32_16x16x4_f32(false, a, false, b, (short)0, c,
                                               false, false);
}

__device__ __forceinline__ v8f v8zero() {
  v8f z = {0.f, 0.f, 0.f, 0.f, 0.f, 0.f, 0.f, 0.f};
  return z;
}

// ---------------------------------------------------------------------------
// LDS-tiled transpose: xt[b][t][c] = x[b][c][t]
// ---------------------------------------------------------------------------
__global__ __launch_bounds__(256) void k_transpose(const float* __restrict__ x,
                                                   float* __restrict__ xt) {
  __shared__ float tile[32][33];
  const int c0 = blockIdx.x * 32, t0 = blockIdx.y * 32, b = blockIdx.z;
  const int tx = threadIdx.x, ty = threadIdx.y;  // 32 x 8
  const float* xb = x + (size_t)b * C_ * T_;
#pragma unroll
  for (int i = 0; i < 4; ++i)
    tile[ty + 8 * i][tx] = xb[(size_t)(c0 + ty + 8 * i) * T_ + t0 + tx];
  __syncthreads();
  float* xtb = xt + (size_t)b * T_ * C_;
#pragma unroll
  for (int i = 0; i < 4; ++i)
    xtb[(size_t)(t0 + ty + 8 * i) * C_ + c0 + tx] = tile[tx][ty + 8 * i];
}

// ---------------------------------------------------------------------------
// Fold the 4 causal convs (k=1,2,4,8, averaged) into one k=8 causal conv.
// weff stored [d][o][c] so B-fragments are contiguous in c.
// ---------------------------------------------------------------------------
__global__ void k_build_weff(const float* __restrict__ w1, const float* __restrict__ b1,
                             const float* __restrict__ w2, const float* __restrict__ b2,
                             const float* __restrict__ w4, const float* __restrict__ b4,
                             const float* __restrict__ w8, const float* __restrict__ b8,
                             float* __restrict__ weff, float* __restrict__ beff) {
  int idx = blockIdx.x * blockDim.x + threadIdx.x;  // o*256 + c
  if (idx >= C_ * C_) return;
  int o = idx >> 8, c = idx & 255;
#pragma unroll
  for (int d = 0; d < 8; ++d) {
    float s = 0.f;
    if (d < 1) s += w1[(o * C_ + c) * 1 + (0 - d)];
    if (d < 2) s += w2[(o * C_ + c) * 2 + (1 - d)];
    if (d < 4) s += w4[(o * C_ + c) * 4 + (3 - d)];
    s += w8[(o * C_ + c) * 8 + (7 - d)];
    weff[((size_t)d * C_ + o) * C_ + c] = 0.25f * s;
  }
  if (c == 0) beff[o] = 0.25f * (b1[o] + b2[o] + b4[o] + b8[o]);
}

// ---------------------------------------------------------------------------
// DFT matrices. Fr/Fi: [NFP][T] forward (cos, -sin). Cr/Ci: [T][NFP] inverse
// with hermitian weights alpha_f/T. Phase reduced mod 1024 -> exact args.
// Rows/cols f >= 513 are zero.
// ---------------------------------------------------------------------------
__global__ void k_build_dft(float* __restrict__ Fr, float* __restrict__ Fi,
                            float* __restrict__ Cr, float* __restrict__ Ci) {
  int t = blockIdx.x * blockDim.x + threadIdx.x;
  if (t >= T_) return;
  for (int f = 0; f < NFP; ++f) {
    float re = 0.f, im = 0.f, cr = 0.f, ci = 0.f;
    if (f < NF) {
      int ph = (f * t) & (T_ - 1);
      float th = (float)ph * (6.283185307179586f / (float)T_);
      float c = cosf(th), s = sinf(th);
      re = c;
      im = -s;
      float alpha = (f == 0 || f == NF - 1) ? 1.f : 2.f;
      cr = alpha * c * (1.f / (float)T_);
      ci = -alpha * s * (1.f / (float)T_);
    }
    Fr[(size_t)f * T_ + t] = re;
    Fi[(size_t)f * T_ + t] = im;
    Cr[(size_t)t * NFP + f] = cr;
    Ci[(size_t)t * NFP + f] = ci;
  }
}

// ---------------------------------------------------------------------------
// env_rep[b,t,o] = beff[o] + sum_d sum_c weff[d][o][c] * xt[b][t-d][c]
// 32t x 32o per wave; boundary handled branch-free (clamped address * mask).
// ---------------------------------------------------------------------------
__global__ __launch_bounds__(32) void k_conv(const float* __restrict__ xt,
                                             const float* __restrict__ weff,
                                             const float* __restrict__ beff,
                                             float* __restrict__ out) {
  const int lane = threadIdx.x & 31, lm = lane & 15, lh = lane >> 4;
  const int t0 = blockIdx.x * 32, o0 = blockIdx.y * 32, b = blockIdx.z;
  const float* xb = xt + (size_t)b * T_ * C_;
  v8f acc[2][2];
#pragma unroll
  for (int ti = 0; ti < 2; ++ti)
#pragma unroll
    for (int oi = 0; oi < 2; ++oi) acc[ti][oi] = v8zero();

  for (int d = 0; d < 8; ++d) {
    const int ta0 = t0 + lm - d;           // only tile ti=0 of block x=0 can underflow
    const float m0 = (ta0 >= 0) ? 1.f : 0.f;
    const size_t r0 = (size_t)(ta0 >= 0 ? ta0 : 0) * C_;
    const size_t r1 = (size_t)(t0 + 16 + lm - d) * C_;
    const float* wd0 = weff + ((size_t)d * C_ + (o0 + lm)) * C_;
    const float* wd1 = wd0 + 16 * C_;
    for (int c0 = 0; c0 < C_; c0 += 4) {
      const int c = c0 + 2 * lh;
      v2f a0, a1, b0, b1;
      a0[0] = xb[r0 + c] * m0;
      a0[1] = xb[r0 + c + 1] * m0;
      a1[0] = xb[r1 + c];
      a1[1] = xb[r1 + c + 1];
      b0[0] = wd0[c];
      b0[1] = wd0[c + 1];
      b1[0] = wd1[c];
      b1[1] = wd1[c + 1];
      acc[0][0] = wmma4(a0, b0, acc[0][0]);
      acc[0][1] = wmma4(a0, b1, acc[0][1]);
      acc[1][0] = wmma4(a1, b0, acc[1][0]);
      acc[1][1] = wmma4(a1, b1, acc[1][1]);
    }
  }
  float* ob = out + (size_t)b * T_ * C_;
#pragma unroll
  for (int ti = 0; ti < 2; ++ti)
#pragma unroll
    for (int oi = 0; oi < 2; ++oi) {
      float bias = beff[o0 + 16 * oi + lm];
#pragma unroll
      for (int e = 0; e < 8; ++e) {
        int t = t0 + 16 * ti + e + 8 * lh;
        ob[(size_t)t * C_ + o0 + 16 * oi + lm] = acc[ti][oi][e] + bias;
      }
    }
}

// ---------------------------------------------------------------------------
// out[b,t,o] = bias[o] + sum_c in[b][t][c] * W[c][o]   (in is [b][t][c])
// 32t x 32o per wave.
// ---------------------------------------------------------------------------
__global__ __launch_bounds__(32) void k_linproj(const float* __restrict__ in,
                                                const float* __restrict__ W,
                                                const float* __restrict__ bias,
                                                float* __restrict__ out) {
  const int lane = threadIdx.x & 31, lm = lane & 15, lh = lane >> 4;
  const int t0 = blockIdx.x * 32, o0 = blockIdx.y * 32, b = blockIdx.z;
  const float* ib = in + (size_t)b * T_ * H_;
  v8f acc[2][2];
#pragma unroll
  for (int ti = 0; ti < 2; ++ti)
#pragma unroll
    for (int oi = 0; oi < 2; ++oi) acc[ti][oi] = v8zero();

  const size_t r0 = (size_t)(t0 + lm) * H_;
  const size_t r1 = r0 + (size_t)16 * H_;
  for (int c0 = 0; c0 < H_; c0 += 4) {
    const int c = c0 + 2 * lh;
    v2f a0, a1, b0, b1;
    a0[0] = ib[r0 + c];
    a0[1] = ib[r0 + c + 1];
    a1[0] = ib[r1 + c];
    a1[1] = ib[r1 + c + 1];
    const size_t wr = (size_t)c * H_ + o0 + lm;
    b0[0] = W[wr];
    b0[1] = W[wr + H_];
    b1[0] = W[wr + 16];
    b1[1] = W[wr + 16 + H_];
    acc[0][0] = wmma4(a0, b0, acc[0][0]);
    acc[0][1] = wmma4(a0, b1, acc[0][1]);
    acc[1][0] = wmma4(a1, b0, acc[1][0]);
    acc[1][1] = wmma4(a1, b1, acc[1][1]);
  }
#pragma unroll
  for (int ti = 0; ti < 2; ++ti)
#pragma unroll
    for (int oi = 0; oi < 2; ++oi) {
      float bv = bias[o0 + 16 * oi + lm];
#pragma unroll
      for (int e = 0; e < 8; ++e) {
        int t = t0 + 16 * ti + e + 8 * lh;
        out[((size_t)b * T_ + t) * H_ + o0 + 16 * oi + lm] =
            acc[ti][oi][e] + bv;
      }
    }
}

// ---------------------------------------------------------------------------
// Flash attention, one wave per (b, head, 16-row Q tile).
// S^T = K @ Q^T puts softmax rows (t) on the lane axis; O^T = V^T @ P^T with
// P^T register-resident (half-wave shuffles build B-fragments).
// ---------------------------------------------------------------------------
__global__ __launch_bounds__(32) void k_attn(const float* __restrict__ q,
                                             const float* __restrict__ kk,
                                             const float* __restrict__ vv,
                                             float* __restrict__ ao) {
  const int lane = threadIdx.x & 31, lm = lane & 15, lh = lane >> 4;
  const int t0 = blockIdx.x * 16, h = blockIdx.y, b = blockIdx.z;
  const size_t hb = (size_t)h * DH;
  const float* qb = q + (size_t)b * T_ * H_;
  const float* kb = kk + (size_t)b * T_ * H_;
  const float* vb = vv + (size_t)b * T_ * H_;

  v2f qf[16];
#pragma unroll
  for (int ds = 0; ds < 16; ++ds) {
    size_t base = (size_t)(t0 + lm) * H_ + hb + 4 * ds + 2 * lh;
    qf[ds][0] = qb[base];
    qf[ds][1] = qb[base + 1];
  }

  float m_run = -3.0e38f, l_run = 0.f;
  v8f oacc[4];
#pragma unroll
  for (int dc = 0; dc < 4; ++dc) oacc[dc] = v8zero();

  for (int s0 = 0; s0 < T_; s0 += 16) {
    v8f sacc = v8zero();
#pragma unroll
    for (int ds = 0; ds < 16; ++ds) {
      size_t base = (size_t)(s0 + lm) * H_ + hb + 4 * ds + 2 * lh;
      v2f af;
      af[0] = kb[base];
      af[1] = kb[base + 1];
      sacc = wmma4(af, qf[ds], sacc);
    }
    float p[8];
    float tmax = -3.0e38f;
#pragma unroll
    for (int e = 0; e < 8; ++e) {
      p[e] = sacc[e] * 0.125f;  // 1/sqrt(64)
      tmax = fmaxf(tmax, p[e]);
    }
    tmax = fmaxf(tmax, __shfl_xor(tmax, 16));
    float m_new = fmaxf(m_run, tmax);
    float corr = __expf(m_run - m_new);
    float tsum = 0.f;
#pragma unroll
    for (int e = 0; e < 8; ++e) {
      p[e] = __expf(p[e] - m_new);
      tsum += p[e];
    }
    tsum += __shfl_xor(tsum, 16);
    l_run = l_run * corr + tsum;
    m_run = m_new;
#pragma unroll
    for (int dc = 0; dc < 4; ++dc)
#pragma unroll
      for (int e = 0; e < 8; ++e) oacc[dc][e] *= corr;

#pragma unroll
    for (int koff = 0; koff < 16; koff += 4) {
      v2f pb;
#pragma unroll
      for (int j = 0; j < 2; ++j) {
        const int sL = koff + j;      // needed by lanes 0-15
        const int sH = koff + 2 + j;  // needed by lanes 16-31
        float sent = lh ? p[sL & 7] : p[sH & 7];
        float got = __shfl_xor(sent, 16);
        float val =
            lh ? ((sH >= 8) ? p[sH & 7] : got) : ((sL < 8) ? p[sL & 7] : got);
        pb[j] = val;
      }
#pragma unroll
      for (int dc = 0; dc < 4; ++dc) {
        size_t base = (size_t)(s0 + koff + 2 * lh) * H_ + hb + dc * 16 + lm;
        v2f af;
        af[0] = vb[base];
        af[1] = vb[base + H_];
        oacc[dc] = wmma4(af, pb, oacc[dc]);
      }
    }
  }
  float inv = 1.f / l_run;
  float* aob = ao + (size_t)b * T_ * H_;
#pragma unroll
  for (int dc = 0; dc < 4; ++dc)
#pragma unroll
    for (int e = 0; e < 8; ++e) {
      int d = dc * 16 + e + 8 * lh;
      aob[(size_t)(t0 + lm) * H_ + hb + d] = oacc[dc][e] * inv;
    }
}

// ---------------------------------------------------------------------------
// Forward DFT as GEMM: xr/xi[b,f,h] = sum_t Fr/Fi[f,t] * x[b,h,t]
// 32f x 32h per wave (B reads original [b][c][t] layout: t contiguous).
// ---------------------------------------------------------------------------
__global__ __launch_bounds__(32) void k_dft(const float* __restrict__ x,
                                            const float* __restrict__ Fr,
                                            const float* __restrict__ Fi,
                                            float* __restrict__ xr,
                                            float* __restrict__ xi) {
  const int lane = threadIdx.x & 31, lm = lane & 15, lh = lane >> 4;
  const int f0 = blockIdx.x * 32, h0 = blockIdx.y * 32, b = blockIdx.z;
  const float* xb = x + (size_t)b * C_ * T_;
  v8f accr[2][2], acci[2][2];
#pragma unroll
  for (int fi = 0; fi < 2; ++fi)
#pragma unroll
    for (int hi = 0; hi < 2; ++hi) {
      accr[fi][hi] = v8zero();
      acci[fi][hi] = v8zero();
    }
  const size_t fr0 = (size_t)(f0 + lm) * T_;
  const size_t fr1 = fr0 + (size_t)16 * T_;
  const size_t xr0 = (size_t)(h0 + lm) * T_;
  const size_t xr1 = xr0 + (size_t)16 * T_;
  for (int t0 = 0; t0 < T_; t0 += 4) {
    const int tt = t0 + 2 * lh;
    v2f ar0, ar1, ai0, ai1, bx0, bx1;
    ar0[0] = Fr[fr0 + tt];
    ar0[1] = Fr[fr0 + tt + 1];
    ar1[0] = Fr[fr1 + tt];
    ar1[1] = Fr[fr1 + tt + 1];
    ai0[0] = Fi[fr0 + tt];
    ai0[1] = Fi[fr0 + tt + 1];
    ai1[0] = Fi[fr1 + tt];
    ai1[1] = Fi[fr1 + tt + 1];
    bx0[0] = xb[xr0 + tt];
    bx0[1] = xb[xr0 + tt + 1];
    bx1[0] = xb[xr1 + tt];
    bx1[1] = xb[xr1 + tt + 1];
    accr[0][0] = wmma4(ar0, bx0, accr[0][0]);
    accr[0][1] = wmma4(ar0, bx1, accr[0][1]);
    accr[1][0] = wmma4(ar1, bx0, accr[1][0]);
    accr[1][1] = wmma4(ar1, bx1, accr[1][1]);
    acci[0][0] = wmma4(ai0, bx0, acci[0][0]);
    acci[0][1] = wmma4(ai0, bx1, acci[0][1]);
    acci[1][0] = wmma4(ai1, bx0, acci[1][0]);
    acci[1][1] = wmma4(ai1, bx1, acci[1][1]);
  }
#pragma unroll
  for (int fi = 0; fi < 2; ++fi)
#pragma unroll
    for (int hi = 0; hi < 2; ++hi)
#pragma unroll
      for (int e = 0; e < 8; ++e) {
        int f = f0 + 16 * fi + e + 8 * lh;
        size_t idx = ((size_t)b * NFP + f) * H_ + h0 + 16 * hi + lm;
        xr[idx] = accr[fi][hi][e];
        xi[idx] = acci[fi][hi][e];
      }
}

// ---------------------------------------------------------------------------
// Per-frequency complex matmul batched over b (M=16=B), 16 x 64o per wave:
//   Yr = Xr@Wr - Xi@Wi + br ;  Yi = Xr@Wi + Xi@Wr + bi
// ---------------------------------------------------------------------------
__global__ __launch_bounds__(32) void k_freqmm(const float* __restrict__ xr,
                                               const float* __restrict__ xi,
                                               const float* __restrict__ Wr,
                                               const float* __restrict__ Wi,
                                               const float* __restrict__ br,
                                               const float* __restrict__ bi,
                                               float* __restrict__ yr,
                                               float* __restrict__ yi) {
  const int lane = threadIdx.x & 31, lm = lane & 15, lh = lane >> 4;
  const int f = blockIdx.x, o0 = blockIdx.y * 64;
  if (f >= NF) {  // zero the padded rows so the inverse-DFT K-loop is clean
#pragma unroll
    for (int oi = 0; oi < 4; ++oi)
#pragma unroll
      for (int e = 0; e < 8; ++e) {
        size_t idx =
            ((size_t)(e + 8 * lh) * NFP + f) * H_ + o0 + 16 * oi + lm;
        yr[idx] = 0.f;
        yi[idx] = 0.f;
      }
    return;
  }
  const float* Wrf = Wr + (size_t)f * H_ * H_;
  const float* Wif = Wi + (size_t)f * H_ * H_;
  v8f accr[4], acci[4];
#pragma unroll
  for (int oi = 0; oi < 4; ++oi) {
    accr[oi] = v8zero();
    acci[oi] = v8zero();
  }
  for (int i0 = 0; i0 < H_; i0 += 4) {
    const int k2 = 2 * lh;
    // prefetch next K-step of the streamed weights (global_prefetch_b8)
    __builtin_prefetch((const void*)(Wrf + (size_t)(i0 + 4 + k2) * H_ + o0 + lm), 0, 0);
    __builtin_prefetch((const void*)(Wif + (size_t)(i0 + 4 + k2) * H_ + o0 + lm), 0, 0);
    size_t ax = ((size_t)lm * NFP + f) * H_ + i0 + k2;
    v2f axr, axi, axin;
    axr[0] = xr[ax];
    axr[1] = xr[ax + 1];
    axi[0] = xi[ax];
    axi[1] = xi[ax + 1];
    axin[0] = -axi[0];
    axin[1] = -axi[1];
    const size_t wb = (size_t)(i0 + k2) * H_ + o0 + lm;
#pragma unroll
    for (int oi = 0; oi < 4; ++oi) {
      v2f bwr, bwi;
      bwr[0] = Wrf[wb + 16 * oi];
      bwr[1] = Wrf[wb + 16 * oi + H_];
      bwi[0] = Wif[wb + 16 * oi];
      bwi[1] = Wif[wb + 16 * oi + H_];
      accr[oi] = wmma4(axr, bwr, accr[oi]);
      accr[oi] = wmma4(axin, bwi, accr[oi]);
      acci[oi] = wmma4(axr, bwi, acci[oi]);
      acci[oi] = wmma4(axi, bwr, acci[oi]);
    }
  }
#pragma unroll
  for (int oi = 0; oi < 4; ++oi) {
    float vbr = br[(size_t)f * H_ + o0 + 16 * oi + lm];
    float vbi = bi[(size_t)f * H_ + o0 + 16 * oi + lm];
#pragma unroll
    for (int e = 0; e < 8; ++e) {
      size_t idx = ((size_t)(e + 8 * lh) * NFP + f) * H_ + o0 + 16 * oi + lm;
      yr[idx] = accr[oi][e] + vbr;
      yi[idx] = acci[oi][e] + vbi;
    }
  }
}

// ---------------------------------------------------------------------------
// Inverse DFT as GEMM accumulated onto entity_time already in d_out:
//   ent[b,t,o] += sum_f Cr[t,f]*yr[b,f,o] + Ci[t,f]*yi[b,f,o]
// 32t x 32o per wave.
// ---------------------------------------------------------------------------
__global__ __launch_bounds__(32) void k_idft(const float* __restrict__ Cr,
                                             const float* __restrict__ Ci,
                                             const float* __restrict__ yr,
                                             const float* __restrict__ yi,
                                             float* __restrict__ ent) {
  const int lane = threadIdx.x & 31, lm = lane & 15, lh = lane >> 4;
  const int t0 = blockIdx.x * 32, o0 = blockIdx.y * 32, b = blockIdx.z;
  v8f acc[2][2];
#pragma unroll
  for (int ti = 0; ti < 2; ++ti)
#pragma unroll
    for (int oi = 0; oi < 2; ++oi) acc[ti][oi] = v8zero();
  const size_t cr0 = (size_t)(t0 + lm) * NFP;
  const size_t cr1 = cr0 + (size_t)16 * NFP;
  for (int f0 = 0; f0 < NFP; f0 += 4) {
    const int k2 = 2 * lh;
    v2f acr0, acr1, aci0, aci1;
    acr0[0] = Cr[cr0 + f0 + k2];
    acr0[1] = Cr[cr0 + f0 + k2 + 1];
    acr1[0] = Cr[cr1 + f0 + k2];
    acr1[1] = Cr[cr1 + f0 + k2 + 1];
    aci0[0] = Ci[cr0 + f0 + k2];
    aci0[1] = Ci[cr0 + f0 + k2 + 1];
    aci1[0] = Ci[cr1 + f0 + k2];
    aci1[1] = Ci[cr1 + f0 + k2 + 1];
    const size_t yb = ((size_t)b * NFP + f0 + k2) * H_ + o0 + lm;
#pragma unroll
    for (int oi = 0; oi < 2; ++oi) {
      v2f byr, byi;
      byr[0] = yr[yb + 16 * oi];
      byr[1] = yr[yb + 16 * oi + H_];
      byi[0] = yi[yb + 16 * oi];
      byi[1] = yi[yb + 16 * oi + H_];
      acc[0][oi] = wmma4(acr0, byr, acc[0][oi]);
      acc[0][oi] = wmma4(aci0, byi, acc[0][oi]);
      acc[1][oi] = wmma4(acr1, byr, acc[1][oi]);
      acc[1][oi] = wmma4(aci1, byi, acc[1][oi]);
    }
  }
#pragma unroll
  for (int ti = 0; ti < 2; ++ti)
#pragma unroll
    for (int oi = 0; oi < 2; ++oi)
#pragma unroll
      for (int e = 0; e < 8; ++e) {
        int t = t0 + 16 * ti + e + 8 * lh;
        size_t idx = ((size_t)b * T_ + t) * H_ + o0 + 16 * oi + lm;
        ent[idx] += acc[ti][oi][e];
      }
}

// ---------------------------------------------------------------------------
extern "C" void kernel_launch(void* const* d_in, const int* in_sizes, int n_in,
                              void* d_out, int out_size, void* d_ws,
                              size_t ws_size, hipStream_t stream) {
  (void)in_sizes;
  (void)n_in;
  (void)out_size;
  (void)ws_size;
  const float* x = (const float*)d_in[0];
  const float* w1 = (const float*)d_in[1];
  const float* b1 = (const float*)d_in[2];
  const float* w2 = (const float*)d_in[3];
  const float* b2 = (const float*)d_in[4];
  const float* w4 = (const float*)d_in[5];
  const float* b4 = (const float*)d_in[6];
  const float* w8 = (const float*)d_in[7];
  const float* b8 = (const float*)d_in[8];
  const float* Wq = (const float*)d_in[9];
  const float* Wk = (const float*)d_in[10];
  const float* Wv = (const float*)d_in[11];
  const float* Wo = (const float*)d_in[12];
  const float* bq = (const float*)d_in[13];
  const float* bk = (const float*)d_in[14];
  const float* bv = (const float*)d_in[15];
  const float* bo = (const float*)d_in[16];
  const float* Fwr = (const float*)d_in[17];
  const float* Fwi = (const float*)d_in[18];
  const float* Fbr = (const float*)d_in[19];
  const float* Fbi = (const float*)d_in[20];

  float* out_env = (float*)d_out;                   // [B,T,256]
  float* out_ent = out_env + (size_t)B_ * T_ * H_;  // [B,T,256]

  // Workspace carve-up (~130 MB of floats).
  float* ws = (float*)d_ws;
  size_t off = 0;
  float* weff = ws + off; off += (size_t)8 * C_ * C_;
  float* beff = ws + off; off += C_;
  float* Fr = ws + off; off += (size_t)NFP * T_;
  float* Fi = ws + off; off += (size_t)NFP * T_;
  float* Cr = ws + off; off += (size_t)T_ * NFP;
  float* Ci = ws + off; off += (size_t)T_ * NFP;
  float* xt = ws + off; off += (size_t)B_ * T_ * C_;
  float* qb = ws + off; off += (size_t)B_ * T_ * H_;
  float* kb = ws + off; off += (size_t)B_ * T_ * H_;
  float* vb = ws + off; off += (size_t)B_ * T_ * H_;
  float* aob = ws + off; off += (size_t)B_ * T_ * H_;
  float* xr = ws + off; off += (size_t)B_ * NFP * H_;
  float* xi = ws + off; off += (size_t)B_ * NFP * H_;
  float* yrb = ws + off; off += (size_t)B_ * NFP * H_;
  float* yib = ws + off; off += (size_t)B_ * NFP * H_;

  // --- precompute ---
  k_build_weff<<<dim3(256), 256, 0, stream>>>(w1, b1, w2, b2, w4, b4, w8, b8,
                                              weff, beff);
  k_build_dft<<<dim3(8), 128, 0, stream>>>(Fr, Fi, Cr, Ci);
  k_transpose<<<dim3(C_ / 32, T_ / 32, B_), dim3(32, 8), 0, stream>>>(x, xt);

  // --- conv branch -> env_rep ---
  k_conv<<<dim3(T_ / 32, C_ / 32, B_), 32, 0, stream>>>(xt, weff, beff,
                                                        out_env);

  // --- attention branch ---
  k_linproj<<<dim3(T_ / 32, H_ / 32, B_), 32, 0, stream>>>(xt, Wq, bq, qb);
  k_linproj<<<dim3(T_ / 32, H_ / 32, B_), 32, 0, stream>>>(xt, Wk, bk, kb);
  k_linproj<<<dim3(T_ / 32, H_ / 32, B_), 32, 0, stream>>>(xt, Wv, bv, vb);
  k_attn<<<dim3(T_ / 16, NHEADS, B_), 32, 0, stream>>>(qb, kb, vb, aob);
  k_linproj<<<dim3(T_ / 32, H_ / 32, B_), 32, 0, stream>>>(aob, Wo, bo,
                                                           out_ent);

  // --- frequency branch, accumulated onto out_ent ---
  k_dft<<<dim3(NFP / 32, H_ / 32, B_), 32, 0, stream>>>(x, Fr, Fi, xr, xi);
  k_freqmm<<<dim3(NFP, H_ / 64), 32, 0, stream>>>(xr, xi, Fwr, Fwi, Fbr, Fbi,
                                                  yrb, yib);
  k_idft<<<dim3(T_ / 32, H_ / 32, B_), 32, 0, stream>>>(Cr, Ci, yrb, yib,
                                                        out_ent);
}